// GraphNorm_43276090474971
// MI455X (gfx1250) — compile-verified
//
#include <hip/hip_runtime.h>
#include <cstdint>
#include <cstddef>

// GraphNorm on MI455X (gfx1250): HBM-bandwidth-bound (~615 MB moved, ~26 us
// floor at 23.3 TB/s). Pipeline:
//   zero -> fused sum/sumsq segment reduce (sorted ids, run-length register
//   accumulation + global f32 atomics) -> per-(g,d) scale/shift fold ->
//   3-deep async global->LDS pipelined normalize stream (ASYNCcnt-tracked,
//   NT hints so the 2 MB scale/shift tables stay L2-resident).

#define FEAT 512
#define NGRAPH 512
#define RED_ROWS 128     // rows per reduce block
#define TILE_ROWS 16     // rows per normalize tile (16*512*4 = 32 KB)
#define NORM_THREADS 256
#define PIPE 3           // async pipeline depth (96 KB LDS of 320 KB/WGP)

typedef float v4f __attribute__((ext_vector_type(4)));

static constexpr float SMALL_NUMBER = 1e-7f;

__global__ void zero_ws_kernel(float* __restrict__ p, int n) {
    int i = blockIdx.x * blockDim.x + threadIdx.x;
    if (i < n) p[i] = 0.0f;
}

// blockDim = (128, 2). Thread x owns feats [4x, 4x+4); thread y strides rows by 2.
// Sorted segment ids -> accumulate in registers, flush on graph change.
__global__ __launch_bounds__(256) void reduce_kernel(
    const float* __restrict__ h, const int* __restrict__ map,
    float* __restrict__ sums, float* __restrict__ sq, float* __restrict__ cnt,
    int N) {
    const int x = threadIdx.x;          // 0..127
    const int y = threadIdx.y;          // 0..1
    const int rbase = blockIdx.x * RED_ROWS;

    v4f s = {0.f, 0.f, 0.f, 0.f};
    v4f q = {0.f, 0.f, 0.f, 0.f};
    int curg = -1;
    float run = 0.f;

    for (int i = y; i < RED_ROWS; i += 2) {
        const int r = rbase + i;
        if (r >= N) break;
        const int g = map[r];
        if (g != curg) {
            if (curg >= 0) {
                float* sp = sums + (size_t)curg * FEAT + x * 4;
                float* qp = sq   + (size_t)curg * FEAT + x * 4;
                atomicAdd(sp + 0, s.x); atomicAdd(sp + 1, s.y);
                atomicAdd(sp + 2, s.z); atomicAdd(sp + 3, s.w);
                atomicAdd(qp + 0, q.x); atomicAdd(qp + 1, q.y);
                atomicAdd(qp + 2, q.z); atomicAdd(qp + 3, q.w);
                if (x == 0) atomicAdd(cnt + curg, run);
                s = (v4f){0.f, 0.f, 0.f, 0.f};
                q = (v4f){0.f, 0.f, 0.f, 0.f};
                run = 0.f;
            }
            curg = g;
        }
        // Streamed exactly once here: non-temporal load.
        const v4f v = __builtin_nontemporal_load(
            reinterpret_cast<const v4f*>(h + (size_t)r * FEAT + x * 4));
        s += v;
        q += v * v;
        run += 1.f;
    }
    if (curg >= 0) {
        float* sp = sums + (size_t)curg * FEAT + x * 4;
        float* qp = sq   + (size_t)curg * FEAT + x * 4;
        atomicAdd(sp + 0, s.x); atomicAdd(sp + 1, s.y);
        atomicAdd(sp + 2, s.z); atomicAdd(sp + 3, s.w);
        atomicAdd(qp + 0, q.x); atomicAdd(qp + 1, q.y);
        atomicAdd(qp + 2, q.z); atomicAdd(qp + 3, q.w);
        if (x == 0) atomicAdd(cnt + curg, run);
    }
}

// out = (h - a*mu) * gamma/(sqrt(var)+eps) + beta  ==  h*scale + shift
// var = E[h^2] - (2a - a^2) * mu^2   (since E[h] = mu)
__global__ void stats_kernel(
    const float* __restrict__ sums, const float* __restrict__ sq,
    const float* __restrict__ cnt,
    const float* __restrict__ alpha, const float* __restrict__ beta,
    const float* __restrict__ gamma,
    float* __restrict__ scale, float* __restrict__ shift) {
    const int i = blockIdx.x * blockDim.x + threadIdx.x;
    if (i >= NGRAPH * FEAT) return;
    const int g = i >> 9;           // / FEAT
    const int d = i & (FEAT - 1);   // % FEAT
    const float c   = fmaxf(cnt[g], 1.0f);
    const float mu  = sums[i] / c;
    const float ex2 = sq[i] / c;
    const float a   = alpha[d];
    const float var = fmaxf(ex2 - (2.0f * a - a * a) * mu * mu, 0.0f);
    const float inv = 1.0f / (sqrtf(var) + SMALL_NUMBER);
    const float sc  = gamma[d] * inv;
    scale[i] = sc;
    shift[i] = fmaf(-a * mu, sc, beta[d]);
}

// Issue one 16x512 tile (32 KB) via async global->LDS b128 copies.
// Each of 256 threads issues 8 x 16B chunks; per-wave ASYNCcnt += 8.
// NT hint: h is stream-once data, keep it out of L2's way.
__device__ __forceinline__ void issue_tile_async(
    const float* __restrict__ h, const int* __restrict__ map,
    int tile, int N, float* sbase, int* gbase, int tid) {
    const int r0 = tile * TILE_ROWS;
    if (tid < TILE_ROWS) {
        int r = r0 + tid;
        if (r >= N) r = N - 1;
        gbase[tid] = map[r];
    }
#pragma unroll
    for (int k = 0; k < 8; ++k) {
        const int c   = tid + k * NORM_THREADS;   // 0..2047 float4 chunks
        const int row = c >> 7;                   // 128 chunks per row
        int r = r0 + row;
        if (r >= N) r = N - 1;                    // clamp: safe duplicate read
        const float* gp = h + (size_t)r * FEAT + (c & 127) * 4;
        const unsigned lp = (unsigned)(uintptr_t)(sbase + c * 4);
        asm volatile("global_load_async_to_lds_b128 %0, %1, off th:TH_LOAD_NT"
                     :: "v"(lp), "v"(gp) : "memory");
    }
}

__global__ __launch_bounds__(NORM_THREADS) void norm_kernel(
    const float* __restrict__ h, const int* __restrict__ map,
    const float* __restrict__ scale, const float* __restrict__ shift,
    float* __restrict__ out, int N, int ntiles) {
    __shared__ float sbuf[PIPE][TILE_ROWS * FEAT];   // 96 KB ring
    __shared__ int   sgid[PIPE][TILE_ROWS];

    const int tid = threadIdx.x;
    const int stride = gridDim.x;
    const int t0 = blockIdx.x;
    if (t0 >= ntiles) return;

    // Prologue: stage up to two leading tiles.
    issue_tile_async(h, map, t0, N, &sbuf[0][0], &sgid[0][0], tid);
    if (t0 + stride < ntiles)
        issue_tile_async(h, map, t0 + stride, N, &sbuf[1][0], &sgid[1][0], tid);

    int slot = 0;
    for (int t = t0; t < ntiles; t += stride) {
        const int t2 = t + 2 * stride;
        const int slot2 = (slot + 2) % PIPE;
        if (t2 < ntiles)
            issue_tile_async(h, map, t2, N, &sbuf[slot2][0], &sgid[slot2][0], tid);

        // Retire the oldest tile's 8 async ops (in-order completion);
        // up to two leading tiles (16 ops) may remain in flight.
        if (t2 < ntiles)               asm volatile("s_wait_asynccnt 0x10" ::: "memory");
        else if (t + stride < ntiles)  asm volatile("s_wait_asynccnt 0x8"  ::: "memory");
        else                           asm volatile("s_wait_asynccnt 0x0"  ::: "memory");
        __syncthreads();

        const int r0 = t * TILE_ROWS;
        const v4f* sb4 = reinterpret_cast<const v4f*>(&sbuf[slot][0]);
#pragma unroll
        for (int k = 0; k < 8; ++k) {
            const int c   = tid + k * NORM_THREADS;
            const int row = c >> 7;
            const int col = c & 127;
            const int r = r0 + row;
            if (r < N) {
                const int g = sgid[slot][row];
                const v4f v  = sb4[c];
                const v4f sc = *reinterpret_cast<const v4f*>(scale + (size_t)g * FEAT + col * 4);
                const v4f sh = *reinterpret_cast<const v4f*>(shift + (size_t)g * FEAT + col * 4);
                v4f o;
                o.x = fmaf(v.x, sc.x, sh.x);
                o.y = fmaf(v.y, sc.y, sh.y);
                o.z = fmaf(v.z, sc.z, sh.z);
                o.w = fmaf(v.w, sc.w, sh.w);
                // Write-once output: non-temporal store.
                __builtin_nontemporal_store(
                    o, reinterpret_cast<v4f*>(out + (size_t)r * FEAT + col * 4));
            }
        }
        __syncthreads();   // buffer `slot` may be refilled at iteration t+PIPE-2
        slot = (slot + 1) % PIPE;
    }
}

extern "C" void kernel_launch(void* const* d_in, const int* in_sizes, int n_in,
                              void* d_out, int out_size, void* d_ws, size_t ws_size,
                              hipStream_t stream) {
    const float* h     = (const float*)d_in[0];
    const int*   map   = (const int*)d_in[1];
    const float* alpha = (const float*)d_in[2];
    const float* beta  = (const float*)d_in[3];
    const float* gamma = (const float*)d_in[4];
    float* out = (float*)d_out;
    const int N = in_sizes[1];

    // Workspace layout (floats): sums[G*D] | sq[G*D] | cnt[G] | scale[G*D] | shift[G*D]
    const size_t GD = (size_t)NGRAPH * FEAT;
    float* ws    = (float*)d_ws;
    float* sums  = ws;
    float* sq    = ws + GD;
    float* cnt   = ws + 2 * GD;
    float* scale = ws + 2 * GD + NGRAPH;
    float* shift = scale + GD;

    const int zn = (int)(2 * GD + NGRAPH);
    zero_ws_kernel<<<(zn + 255) / 256, 256, 0, stream>>>(ws, zn);

    dim3 rblk(128, 2);
    reduce_kernel<<<(N + RED_ROWS - 1) / RED_ROWS, rblk, 0, stream>>>(
        h, map, sums, sq, cnt, N);

    const int sn = (int)GD;
    stats_kernel<<<(sn + 255) / 256, 256, 0, stream>>>(
        sums, sq, cnt, alpha, beta, gamma, scale, shift);

    const int ntiles = (N + TILE_ROWS - 1) / TILE_ROWS;
    int blocks = ntiles < 1536 ? ntiles : 1536;
    norm_kernel<<<blocks, NORM_THREADS, 0, stream>>>(
        h, map, scale, shift, out, N, ntiles);
}